// HMLSTM_25323127177935
// MI455X (gfx1250) — compile-verified
//
#include <hip/hip_runtime.h>

// HMLSTM forward on gfx1250: persistent kernel, fp32 WMMA 16x16x4, grid barrier.
// T=512, B=64, DIN=256, D=512, L=3, G=4*D+1=2049 (padded to 2064 columns).

typedef __attribute__((ext_vector_type(2))) float v2f;
typedef __attribute__((ext_vector_type(8))) float v8f;

#define NBLK 33          // 32 GEMM blocks + 1 boundary-gate block
#define NTHR 128         // 4 waves of 32
#define GPAD 2064        // 2049 padded to 16
#define ROWF (GPAD * 2)  // floats per k-pair row in packed weights = 4128

__device__ __forceinline__ v8f wmma4(v2f a, v2f b, v8f c) {
  // D = A(16x4 f32) x B(4x16 f32) + C(16x16 f32)
  return __builtin_amdgcn_wmma_f32_16x16x4_f32(false, a, false, b, (short)0, c,
                                               false, false);
}

// ---- device-wide generation barrier (33 blocks are co-resident) ----
__device__ __forceinline__ void gbar(unsigned* cnt, unsigned* gen) {
  __syncthreads();
  if (threadIdx.x == 0) {
    __threadfence();  // release this block's global writes (agent scope)
    unsigned g = __hip_atomic_load(gen, __ATOMIC_RELAXED, __HIP_MEMORY_SCOPE_AGENT);
    unsigned prev = __hip_atomic_fetch_add(cnt, 1u, __ATOMIC_ACQ_REL,
                                           __HIP_MEMORY_SCOPE_AGENT);
    if (prev == NBLK - 1u) {
      __hip_atomic_store(cnt, 0u, __ATOMIC_RELAXED, __HIP_MEMORY_SCOPE_AGENT);
      __hip_atomic_store(gen, g + 1u, __ATOMIC_RELEASE, __HIP_MEMORY_SCOPE_AGENT);
    } else {
      while (__hip_atomic_load(gen, __ATOMIC_ACQUIRE,
                               __HIP_MEMORY_SCOPE_AGENT) == g)
        __builtin_amdgcn_s_sleep(1);
    }
    __threadfence();  // acquire other blocks' writes
  }
  __syncthreads();
}

// ---- one-time repack: WTp[k/2][g*2 + (k&1)] = Wcat[g][k], Wcat=[U11|W10|U21]
__device__ void pack_layer(float* __restrict__ WTp, const float* __restrict__ U11,
                           const float* __restrict__ W10,
                           const float* __restrict__ U21, int Kb, int Ktot) {
  const int total = Ktot * GPAD;
  for (int idx = blockIdx.x * NTHR + threadIdx.x; idx < total;
       idx += NBLK * NTHR) {
    int g = idx % GPAD;
    int k = idx / GPAD;
    float v = 0.f;
    if (g < 2049) {
      if (k < 512)            v = U11[(size_t)g * 512 + k];
      else if (k < 512 + Kb)  v = W10[(size_t)g * Kb + (k - 512)];
      else                    v = U21[(size_t)g * 512 + (k - 512 - Kb)];
    }
    WTp[(size_t)(k >> 1) * ROWF + g * 2 + (k & 1)] = v;
  }
}

// ---- WMMA K-segment: s += Wcat[:, kwt:kwt+klen] @ (sc * act) ----
__device__ __forceinline__ void seg(const float* __restrict__ WTp, int kwt,
                                    const float* __restrict__ act, float sc,
                                    int klen, int col /* d0+m (f2 units) */,
                                    int h2, int n, v8f& aF, v8f& aI, v8f& aO,
                                    v8f& aG) {
#pragma unroll 2
  for (int kk = 0; kk < klen; kk += 4) {
    const int ka = kwt + kk;
    const v2f* row = (const v2f*)(WTp + (size_t)((ka >> 1) + h2) * ROWF);
    const float* ap = act + (kk + 2 * h2) * 64 + n;
    v2f b;                      // B frag: K = j + 2*h2 (ISA B layout)
    b.x = ap[0] * sc;
    b.y = ap[64] * sc;
    v2f a0 = row[col];          // gate f rows
    v2f a1 = row[col + 512];    // gate i rows
    v2f a2 = row[col + 1024];   // gate o rows
    v2f a3 = row[col + 1536];   // gate g rows
    aF = wmma4(a0, b, aF);
    aI = wmma4(a1, b, aI);
    aO = wmma4(a2, b, aO);
    aG = wmma4(a3, b, aG);
  }
}

__device__ __forceinline__ float sigmoidf(float x) {
  return 1.f / (1.f + expf(-x));
}

// ---- one layer stage on GEMM blocks 0..31 ----
__device__ void gemm_stage(int t, int l, const float* __restrict__ WTp,
                           const float* __restrict__ bias,
                           const float* __restrict__ x, float* out, float* cbuf,
                           const float* zbuf, int Kb, bool hasTop) {
  const int tid = threadIdx.x;
  const int wid = tid >> 5;   // n-tile 0..3
  const int lane = tid & 31;
  const int m = lane & 15;
  const int h2 = lane >> 4;
  const int n = wid * 16 + m;
  const int d0 = blockIdx.x * 16;
  const int col = d0 + m;

  const int cur = t & 1, prv = cur ^ 1;
  const float zb = (l == 0) ? 1.f : zbuf[cur * 192 + (l - 1) * 64 + n];
  const float zl = (t == 0) ? 0.f : zbuf[prv * 192 + l * 64 + n];

  v8f aF = {}, aI = {}, aO = {}, aG = {};

  if (t > 0) {  // U11 @ h_prev[l]
    const float* hp = out + ((size_t)(t - 1) * 3 + l) * (512 * 64);
    seg(WTp, 0, hp, 1.f, 512, col, h2, n, aF, aI, aO, aG);
  }
  {  // W10 @ (zb * h_below)
    const float* hb = (l == 0) ? (x + (size_t)t * 256 * 64)
                               : (out + ((size_t)t * 3 + (l - 1)) * (512 * 64));
    seg(WTp, 512, hb, zb, Kb, col, h2, n, aF, aI, aO, aG);
  }
  if (hasTop && t > 0) {  // U21 @ (zl * h_prev[l+1])
    const float* ht = out + ((size_t)(t - 1) * 3 + (l + 1)) * (512 * 64);
    seg(WTp, 512 + Kb, ht, zl, 512, col, h2, n, aF, aI, aO, aG);
  }

  // epilogue: gates -> c,h (register-local; C frag element j is row d0+j+8*h2)
#pragma unroll
  for (int j = 0; j < 8; ++j) {
    const int d = d0 + j + 8 * h2;
    const float sf = aF[j] + bias[d];
    const float si = aI[j] + bias[512 + d];
    const float so = aO[j] + bias[1024 + d];
    const float sg = aG[j] + bias[1536 + d];
    const float f = sigmoidf(sf);
    const float i = sigmoidf(si);
    const float o = sigmoidf(so);
    const float g = tanhf(sg);
    const size_t cix = ((size_t)l * 512 + d) * 64 + n;
    const float c_old = (t == 0) ? 0.f : cbuf[cix];
    const float h_old =
        (t == 0) ? 0.f : out[(((size_t)(t - 1) * 3 + l) * 512 + d) * 64 + n];
    const float ig = i * g;
    const float cu = f * c_old + ig;  // UPDATE
    const float c_new = zl * ig + (1.f - zl) * ((1.f - zb) * c_old + zb * cu);
    const float tc = tanhf(c_new);
    const float otc = o * tc;
    const float h_new = zl * otc + (1.f - zl) * ((1.f - zb) * h_old + zb * otc);
    cbuf[cix] = c_new;
    out[(((size_t)t * 3 + l) * 512 + d) * 64 + n] = h_new;
  }
}

// ---- boundary-gate row (g = 2048) on block 32: z_new = (s > 0) ----
__device__ void z_stage(int t, int l, const float* __restrict__ WTp,
                        const float* __restrict__ bias,
                        const float* __restrict__ x, const float* out,
                        float* zbuf, int Kb, bool hasTop, float* red) {
  const int tid = threadIdx.x;
  const int n = tid & 63;
  const int half = tid >> 6;  // split K two ways
  const int cur = t & 1, prv = cur ^ 1;
  const float zb = (l == 0) ? 1.f : zbuf[cur * 192 + (l - 1) * 64 + n];
  const float zl = (t == 0) ? 0.f : zbuf[prv * 192 + l * 64 + n];
  float acc = 0.f;
  auto zseg = [&](int kwt, const float* act, float sc, int klen) {
    const int kb = half * (klen >> 1), ke = kb + (klen >> 1);
    for (int k = kb; k < ke; ++k) {
      const int ka = kwt + k;
      const float w = WTp[(size_t)(ka >> 1) * ROWF + 4096 + (ka & 1)];
      acc += w * act[k * 64 + n] * sc;
    }
  };
  if (t > 0) zseg(0, out + ((size_t)(t - 1) * 3 + l) * (512 * 64), 1.f, 512);
  const float* hb = (l == 0) ? (x + (size_t)t * 256 * 64)
                             : (out + ((size_t)t * 3 + (l - 1)) * (512 * 64));
  zseg(512, hb, zb, Kb);
  if (hasTop && t > 0)
    zseg(512 + Kb, out + ((size_t)(t - 1) * 3 + (l + 1)) * (512 * 64), zl, 512);
  red[tid] = acc;
  __syncthreads();
  if (tid < 64) {
    const float s = red[tid] + red[tid + 64] + bias[2048];
    zbuf[cur * 192 + l * 64 + tid] = (s > 0.f) ? 1.f : 0.f;
  }
  __syncthreads();
}

__global__ __launch_bounds__(NTHR, 1) void hmlstm_persistent(
    const float* __restrict__ x, const float* __restrict__ W10_0,
    const float* __restrict__ U11_0, const float* __restrict__ U21_0,
    const float* __restrict__ b_0, const float* __restrict__ W10_1,
    const float* __restrict__ U11_1, const float* __restrict__ U21_1,
    const float* __restrict__ b_1, const float* __restrict__ W10_2,
    const float* __restrict__ U11_2, const float* __restrict__ b_2, float* out,
    float* ws) {
  __shared__ float red[NTHR];
  unsigned* cnt = (unsigned*)ws;      // ws[0..63]: barrier words (memset by host)
  unsigned* gen = (unsigned*)ws + 1;
  float* WTp0 = ws + 64;              // 640 * 4128
  float* WTp1 = WTp0 + 640 * ROWF;    // 768 * 4128
  float* WTp2 = WTp1 + 768 * ROWF;    // 512 * 4128
  float* cbuf = WTp2 + 512 * ROWF;    // 3*512*64
  float* zbuf = cbuf + 3 * 512 * 64;  // 2*3*64

  pack_layer(WTp0, U11_0, W10_0, U21_0, 256, 1280);
  pack_layer(WTp1, U11_1, W10_1, U21_1, 512, 1536);
  pack_layer(WTp2, U11_2, W10_2, nullptr, 512, 1024);
  gbar(cnt, gen);

  for (int t = 0; t < 512; ++t) {
    for (int l = 0; l < 3; ++l) {
      const float* WTp = (l == 0) ? WTp0 : ((l == 1) ? WTp1 : WTp2);
      const float* bias = (l == 0) ? b_0 : ((l == 1) ? b_1 : b_2);
      const int Kb = (l == 0) ? 256 : 512;
      const bool hasTop = (l < 2);
      if (blockIdx.x < 32)
        gemm_stage(t, l, WTp, bias, x, out, cbuf, zbuf, Kb, hasTop);
      else
        z_stage(t, l, WTp, bias, x, out, zbuf, Kb, hasTop, red);
      gbar(cnt, gen);
    }
  }
}

extern "C" void kernel_launch(void* const* d_in, const int* in_sizes, int n_in,
                              void* d_out, int out_size, void* d_ws,
                              size_t ws_size, hipStream_t stream) {
  (void)in_sizes; (void)n_in; (void)out_size; (void)ws_size;
  const float* x     = (const float*)d_in[0];
  const float* W10_0 = (const float*)d_in[1];
  const float* U11_0 = (const float*)d_in[2];
  const float* U21_0 = (const float*)d_in[3];
  const float* b_0   = (const float*)d_in[4];
  const float* W10_1 = (const float*)d_in[5];
  const float* U11_1 = (const float*)d_in[6];
  const float* U21_1 = (const float*)d_in[7];
  const float* b_1   = (const float*)d_in[8];
  const float* W10_2 = (const float*)d_in[9];
  const float* U11_2 = (const float*)d_in[10];
  const float* b_2   = (const float*)d_in[11];

  hipMemsetAsync(d_ws, 0, 256, stream);  // barrier counter/generation
  hmlstm_persistent<<<dim3(NBLK), dim3(NTHR), 0, stream>>>(
      x, W10_0, U11_0, U21_0, b_0, W10_1, U11_1, U21_1, b_1, W10_2, U11_2, b_2,
      (float*)d_out, (float*)d_ws);
}